// RoPE2D_6717328850967
// MI455X (gfx1250) — compile-verified
//
#include <hip/hip_runtime.h>
#include <cstdint>

typedef __attribute__((ext_vector_type(4))) float v4f;

// 2*pi/10, float32
#define ROPE_THETA 0.6283185307179586f

// Layout: x[t][j][d], t in [0,T), j in {0,1}, d in [0,4096). Row stride = 8192 floats.
// One 256-thread block (8 wave32) processes one full row t:
//   - stages the 16 KB x0 row and 16 KB x1 row into LDS with 8 async b128
//     copies per thread (CDNA5 ASYNCcnt path; INST_OFFSET advances both the
//     LDS and the global address per the ISA's async addressing rule),
//   - waits on ASYNCcnt (each thread consumes only its own staged bytes, so
//     no workgroup barrier is required),
//   - computes sin/cos once per thread via a single shared range reduction,
//   - rotates and writes back with non-temporal b128 stores (512 MiB streamed
//     once > 192 MB L2, so don't displace cache lines).
__global__ __launch_bounds__(256) void rope2d_async_kernel(
    const float* __restrict__ x, float* __restrict__ out, int T) {
  __shared__ float lds[8192];  // 32 KB: [0..4095]=x0 row, [4096..8191]=x1 row

  const int tid = threadIdx.x;   // 0..255
  const int t   = blockIdx.x;    // row index
  if (t >= T) return;

  const long long rowBase = (long long)t * 8192ll;
  const int d0 = tid << 2;       // this thread's base float index within the row

  const float* g0 = x + rowBase + d0;           // x[t][0][d0..]
  const float* g1 = x + rowBase + 4096 + d0;    // x[t][1][d0..]

  // LDS byte addresses (low 32 bits of the flat shared-aperture address == LDS offset)
  const unsigned l0 = (unsigned)(uintptr_t)&lds[d0];
  const unsigned l1 = (unsigned)(uintptr_t)&lds[4096 + d0];

  // CDNA5 async global->LDS b128 copies. The 24-bit instruction offset is
  // added to BOTH the LDS destination and the global source address
  // (CDNA5 ISA), so one LDS-addr VGPR + one global-addr
  // VGPR pair serves all 4 slices of each half (slice stride = 1024 floats).
#define ASYNC_CP(LDSA, GPTR, OFS)                                          \
  asm volatile("global_load_async_to_lds_b128 %0, %1, off offset:" #OFS    \
               :: "v"(LDSA), "v"(GPTR) : "memory")
  ASYNC_CP(l0, g0, 0);
  ASYNC_CP(l0, g0, 4096);
  ASYNC_CP(l0, g0, 8192);
  ASYNC_CP(l0, g0, 12288);
  ASYNC_CP(l1, g1, 0);
  ASYNC_CP(l1, g1, 4096);
  ASYNC_CP(l1, g1, 8192);
  ASYNC_CP(l1, g1, 12288);
#undef ASYNC_CP
  asm volatile("s_wait_asynccnt 0" ::: "memory");

  // One shared range reduction for both sin and cos (angle matches the
  // reference's float32 evaluation order: float(t) * (2*pi/10)).
  float s, c;
  sincosf((float)t * ROPE_THETA, &s, &c);

#pragma unroll
  for (int i = 0; i < 4; ++i) {
    const int d = (i << 10) + d0;               // slice stride = 1024 floats
    v4f a = *(const v4f*)&lds[d];               // x0
    v4f b = *(const v4f*)&lds[4096 + d];        // x1

    v4f o0 = c * a - s * b;                     // c*x0 - s*x1
    v4f o1 = s * a + c * b;                     // s*x0 + c*x1

    __builtin_nontemporal_store(o0, (v4f*)(out + rowBase + d));
    __builtin_nontemporal_store(o1, (v4f*)(out + rowBase + 4096 + d));
  }
}

extern "C" void kernel_launch(void* const* d_in, const int* in_sizes, int n_in,
                              void* d_out, int out_size, void* d_ws, size_t ws_size,
                              hipStream_t stream) {
  const float* x = (const float*)d_in[0];
  float* out = (float*)d_out;
  const int n = in_sizes[0];          // T * 2 * 4096
  const int T = n / 8192;
  dim3 grid((unsigned)T), block(256);
  rope2d_async_kernel<<<grid, block, 0, stream>>>(x, out, T);
}